// GCN_layers_3521873183316
// MI455X (gfx1250) — compile-verified
//
#include <hip/hip_runtime.h>
#include <math.h>

#define B_ 128
#define N_ 256
#define D_ 768
#define LN_EPS 1e-6f

typedef __attribute__((ext_vector_type(2))) float v2f;
typedef __attribute__((ext_vector_type(8))) float v8f;

// ---------------------------------------------------------------------------
// Kernel 1: per-batch active length L[b] = sum(mask[b,:]) and tZ[d]=tanh(b1[d])
// ---------------------------------------------------------------------------
__global__ void prep_kernel(const float* __restrict__ mask,
                            const float* __restrict__ b1,
                            float* __restrict__ Lcnt,
                            float* __restrict__ tZ) {
    int t = threadIdx.x;
    if (t < D_) tZ[t] = tanhf(b1[t]);
    if (t < B_) {
        float s = 0.f;
        #pragma unroll 8
        for (int n = 0; n < N_; ++n) s += mask[t * N_ + n];
        Lcnt[t] = s;
    }
}

// ---------------------------------------------------------------------------
// Kernel 2: S1[b,d] = sum_n mask[b,n] * rep[b,n,d]   (the only full read #1)
// grid = (D/256, B), block = 256. Coalesced 1KB lines per step.
// ---------------------------------------------------------------------------
__global__ void masked_sum_kernel(const float* __restrict__ rep,
                                  const float* __restrict__ mask,
                                  float* __restrict__ S1) {
    int d = blockIdx.x * blockDim.x + threadIdx.x;   // 0..D-1
    int b = blockIdx.y;
    const float* rp = rep + (size_t)b * N_ * D_ + d;
    const float* mp = mask + b * N_;
    float acc = 0.f;
    #pragma unroll 4
    for (int n = 0; n < N_; ++n)
        acc += mp[n] * rp[(size_t)n * D_];
    S1[b * D_ + d] = acc;
}

// ---------------------------------------------------------------------------
// Kernel 3: Out[m,n] = (scale ? scale[m] : 1) * tanh( (A @ W)[m,n] + bias[n] )
// A: (128, 768) row-major, W: (768, 768) [k][n] row-major.
// Full-fp32 WMMA: V_WMMA_F32_16X16X4_F32, one 16x16 tile per wave.
// Block = 8 waves (one M-tile each), sharing the same N-tile of W.
// Layouts per CDNA5 ISA 7.12.2:
//   A 16x4 : VGPR v, lanes L: element (M=L%16, K = 2*(L/16) + v)
//   B 4x16 : VGPR v, lanes L: element (K = 2*(L/16) + v, N=L%16)
//   C 16x16: VGPR r, lanes L: element (M = r + 8*(L/16),  N=L%16)
// ---------------------------------------------------------------------------
__global__ void gemm_tanh_kernel(const float* __restrict__ A,
                                 const float* __restrict__ W,
                                 const float* __restrict__ bias,
                                 const float* __restrict__ scale,
                                 float* __restrict__ Out) {
    const int lane = threadIdx.x & 31;
    const int wave = threadIdx.x >> 5;   // 0..7 -> M tile
    const int m0 = wave * 16;
    const int n0 = blockIdx.x * 16;      // N tile
    const int lm = lane & 15;
    const int hi = lane >> 4;            // 0/1: K sub-pair

    const float* arow = A + (size_t)(m0 + lm) * D_;
    v8f c = {};
    #pragma unroll 4
    for (int k0 = 0; k0 < D_; k0 += 4) {
        const int ka = k0 + hi * 2;
        // A: two contiguous K values for this lane (8B-aligned vector load)
        const float2 a2 = *(const float2*)(arow + ka);
        v2f a; a.x = a2.x; a.y = a2.y;
        // B: same two K rows of W at column n0+lm
        v2f b;
        b.x = W[(size_t)ka * D_ + n0 + lm];
        b.y = W[(size_t)(ka + 1) * D_ + n0 + lm];
        c = __builtin_amdgcn_wmma_f32_16x16x4_f32(
                /*neg_a=*/false, a, /*neg_b=*/false, b,
                /*c_mod=*/(short)0, c, /*reuse_a=*/false, /*reuse_b=*/false);
    }

    const int col = n0 + lm;
    const float bb = bias[col];
    #pragma unroll
    for (int r = 0; r < 8; ++r) {
        const int row = m0 + r + hi * 8;
        float v = tanhf(c[r] + bb);
        if (scale) v *= scale[row];
        Out[(size_t)row * D_ + col] = v;
    }
}

// ---------------------------------------------------------------------------
// Kernel 4: out[b,n,:] = LN( rep[b,n,:] + (mask ? T2[b,:] : tZ[:]) )
// One block per (b,n) row; 256 threads x 3 elements; exact two-pass LN.
// ---------------------------------------------------------------------------
__device__ __forceinline__ float block_sum(float v, float* lds) {
    __syncthreads();                       // protect lds reuse across calls
    #pragma unroll
    for (int o = 16; o; o >>= 1) v += __shfl_xor(v, o, 32);
    const int wave = threadIdx.x >> 5, lane = threadIdx.x & 31;
    if (lane == 0) lds[wave] = v;
    __syncthreads();
    if (wave == 0) {
        float r = (lane < 8) ? lds[lane] : 0.f;
        #pragma unroll
        for (int o = 4; o; o >>= 1) r += __shfl_xor(r, o, 32);
        if (lane == 0) lds[0] = r;
    }
    __syncthreads();
    return lds[0];
}

__global__ void final_ln_kernel(const float* __restrict__ rep,
                                const float* __restrict__ mask,
                                const float* __restrict__ T2,
                                const float* __restrict__ tZ,
                                const float* __restrict__ gamma,
                                const float* __restrict__ beta,
                                float* __restrict__ out) {
    __shared__ float lds[8];
    const int row = blockIdx.x;            // 0 .. B*N-1
    const int b   = row >> 8;              // N_ == 256
    const int t   = threadIdx.x;

    const bool act = (mask[row] != 0.f);   // mask is exactly 0.0 / 1.0
    const float* xr = rep + (size_t)row * D_;
    const float* t2 = T2 + (size_t)b * D_;

    float x[3];
    float s = 0.f;
    #pragma unroll
    for (int j = 0; j < 3; ++j) {
        const int d = t + j * 256;
        const float add = act ? t2[d] : tZ[d];
        x[j] = xr[d] + add;
        s += x[j];
    }
    const float mu = block_sum(s, lds) * (1.f / (float)D_);

    float vs = 0.f;
    #pragma unroll
    for (int j = 0; j < 3; ++j) {
        const float dv = x[j] - mu;
        vs += dv * dv;
    }
    const float var = block_sum(vs, lds) * (1.f / (float)D_);
    const float inv = rsqrtf(var + LN_EPS);

    #pragma unroll
    for (int j = 0; j < 3; ++j) {
        const int d = t + j * 256;
        const float v = gamma[d] * (x[j] - mu) * inv + beta[d];
        __builtin_nontemporal_store(v, &out[(size_t)row * D_ + d]);
    }
}

// ---------------------------------------------------------------------------
extern "C" void kernel_launch(void* const* d_in, const int* in_sizes, int n_in,
                              void* d_out, int out_size, void* d_ws, size_t ws_size,
                              hipStream_t stream) {
    const float* rep   = (const float*)d_in[0];
    const float* mask  = (const float*)d_in[1];
    const float* W0    = (const float*)d_in[2];
    const float* b0    = (const float*)d_in[3];
    const float* W1    = (const float*)d_in[4];
    const float* b1    = (const float*)d_in[5];
    const float* gamma = (const float*)d_in[6];
    const float* beta  = (const float*)d_in[7];
    float* out = (float*)d_out;

    float* ws = (float*)d_ws;
    float* S1 = ws;                        // (B, D)
    float* S2 = ws + (size_t)B_ * D_;      // (B, D)
    float* T2 = ws + (size_t)2 * B_ * D_;  // (B, D)
    float* Lc = ws + (size_t)3 * B_ * D_;  // (B)
    float* tZ = Lc + B_;                   // (D)

    prep_kernel<<<1, D_, 0, stream>>>(mask, b1, Lc, tZ);
    masked_sum_kernel<<<dim3(D_ / 256, B_), 256, 0, stream>>>(rep, mask, S1);
    // layer 1: S2 = L[b] * tanh(S1 @ W0 + b0)
    gemm_tanh_kernel<<<D_ / 16, 256, 0, stream>>>(S1, W0, b0, Lc, S2);
    // layer 2: T2 = tanh(S2 @ W1 + b1)
    gemm_tanh_kernel<<<D_ / 16, 256, 0, stream>>>(S2, W1, b1, nullptr, T2);
    final_ln_kernel<<<B_ * N_, 256, 0, stream>>>(rep, mask, T2, tZ, gamma, beta, out);
}